// VQTokenizer_56633438765669
// MI455X (gfx1250) — compile-verified
//
#include <hip/hip_runtime.h>
#include <hip/hip_bf16.h>

// VQ tokenizer for MI455X (gfx1250, wave32).
// Dominant GEMM (8192x8192x512, 68.7 GFLOP) fused with argmin epilogue so the
// 256MB dists matrix is never materialized; fp32 WMMA (16x16x4) keeps argmin
// index fidelity vs the fp32 reference. All tensors are L2-resident (192MB).
// N-blocked by 4: one LDS A-fragment feeds 4 WMMAs; B fragments share one
// base address with immediate offsets (tile stride 32KB < IOFFSET range).

#define T_TOK   8192      // B*C
#define DIM     512       // D
#define KCB     8192      // codebook size
#define FBINS   129
#define CCH     64
#define ZSTRIDE 520       // LDS row stride (floats), breaks bank conflicts, 8B-aligned
#define NBLK    4         // N-tiles per wave iteration

typedef __attribute__((ext_vector_type(2))) float v2f;
typedef __attribute__((ext_vector_type(8))) float v8f;

// ---------------- kernel 1: z = x @ W + b  ([T,129] @ [129,512]) ----------------
__global__ __launch_bounds__(128)
void vq_zproj(const float* __restrict__ x, const float* __restrict__ W,
              const float* __restrict__ bias, float* __restrict__ z) {
    __shared__ float xr[FBINS];
    const int t = blockIdx.x;
    const int tid = threadIdx.x;
    for (int f = tid; f < FBINS; f += 128) xr[f] = x[t * FBINS + f];
    __syncthreads();
    float a0 = 0.f, a1 = 0.f, a2 = 0.f, a3 = 0.f;
    const int d0 = tid;
    for (int f = 0; f < FBINS; ++f) {
        const float xv = xr[f];
        const float* wr = W + f * DIM + d0;
        a0 += xv * wr[0];
        a1 += xv * wr[128];
        a2 += xv * wr[256];
        a3 += xv * wr[384];
    }
    float* zr = z + (size_t)t * DIM + d0;
    zr[0]   = a0 + bias[d0];
    zr[128] = a1 + bias[d0 + 128];
    zr[256] = a2 + bias[d0 + 256];
    zr[384] = a3 + bias[d0 + 384];
}

// ---------------- kernel 2: e2[k] = sum_d codebook[k][d]^2 ----------------
__global__ __launch_bounds__(128)
void vq_cbnorm(const float* __restrict__ cb, float* __restrict__ e2) {
    const int k = blockIdx.x;
    const int tid = threadIdx.x;
    const float* row = cb + (size_t)k * DIM;
    float s = 0.f;
    for (int d = tid; d < DIM; d += 128) { float v = row[d]; s += v * v; }
    #pragma unroll
    for (int m = 16; m >= 1; m >>= 1) s += __shfl_xor(s, m, 32);
    __shared__ float ls[4];
    if ((tid & 31) == 0) ls[tid >> 5] = s;
    __syncthreads();
    if (tid == 0) e2[k] = (ls[0] + ls[1]) + (ls[2] + ls[3]);
}

// ---- kernel 3: fused dots GEMM + argmin + gather + pos_emb + commit partial ----
// Workgroup = 16 tokens, 8 waves. Wave w owns the contiguous N-tile range
// [w*64, w*64+64), processed 4 tiles at a time with register blocking.
__global__ __launch_bounds__(256)
void vq_fused(const float* __restrict__ z, const float* __restrict__ cb,
              const float* __restrict__ e2, const float* __restrict__ pos,
              float* __restrict__ out, float* __restrict__ partial) {
    __shared__ float zt[16 * ZSTRIDE];   // 33,280 B: z tile, padded rows
    __shared__ float wgmin[16][8];
    __shared__ int   wgidx[16][8];
    __shared__ int   fidx[16];
    __shared__ float red[256];

    const int tid = threadIdx.x;
    const int t0  = blockIdx.x * 16;

    // stage z tile (16 tokens x 512) into LDS
    for (int e = tid; e < 16 * DIM; e += 256) {
        const int r = e >> 9, d = e & 511;
        zt[r * ZSTRIDE + d] = z[(size_t)(t0 + r) * DIM + d];
    }
    __syncthreads();

    const int wave = tid >> 5;
    const int lane = tid & 31;
    const int half = lane >> 4;     // K sub-block of the fragment
    const int l16  = lane & 15;     // row (A) / col (B) within tile
    const int koff = half * 2;

    float minv[8];
    int   mini[8];
    #pragma unroll
    for (int v = 0; v < 8; ++v) { minv[v] = 3.402823466e+38f; mini[v] = 0; }

    const float* zrow = zt + l16 * ZSTRIDE + koff;   // A fragment source (LDS)

    for (int nt = wave * 64; nt < wave * 64 + 64; nt += NBLK) {
        const int ncol = (nt << 4) + l16;            // first tile's column
        const float* brow = cb + (size_t)ncol * DIM + koff;
        v8f acc0 = {0.f,0.f,0.f,0.f,0.f,0.f,0.f,0.f};
        v8f acc1 = acc0, acc2 = acc0, acc3 = acc0;
        #pragma unroll 2
        for (int k0 = 0; k0 < DIM; k0 += 4) {
            const v2f a  = *(const v2f*)(zrow + k0);             // reused x4
            const v2f b0 = *(const v2f*)(brow + k0);
            const v2f b1 = *(const v2f*)(brow + k0 + 1 * 16 * DIM);
            const v2f b2 = *(const v2f*)(brow + k0 + 2 * 16 * DIM);
            const v2f b3 = *(const v2f*)(brow + k0 + 3 * 16 * DIM);
            acc0 = __builtin_amdgcn_wmma_f32_16x16x4_f32(false, a, false, b0, (short)0, acc0, false, false);
            acc1 = __builtin_amdgcn_wmma_f32_16x16x4_f32(false, a, false, b1, (short)0, acc1, false, false);
            acc2 = __builtin_amdgcn_wmma_f32_16x16x4_f32(false, a, false, b2, (short)0, acc2, false, false);
            acc3 = __builtin_amdgcn_wmma_f32_16x16x4_f32(false, a, false, b3, (short)0, acc3, false, false);
        }
        // argmin update, ascending column order (tie-break = first min)
        const v8f* accs[NBLK] = { &acc0, &acc1, &acc2, &acc3 };
        #pragma unroll
        for (int j = 0; j < NBLK; ++j) {
            const int   nc  = ncol + j * 16;
            const float e2l = e2[nc];
            #pragma unroll
            for (int v = 0; v < 8; ++v) {
                // z^2 term is per-token constant: argmin-invariant, omitted.
                const float dist = fmaf(-2.0f, (*accs[j])[v], e2l);
                if (dist < minv[v]) { minv[v] = dist; mini[v] = nc; }
            }
        }
    }

    // reduce across the 16 lanes holding one row (C layout: lane<16 row v, lane>=16 row v+8)
    #pragma unroll
    for (int v = 0; v < 8; ++v) {
        float mv = minv[v];
        int   mi = mini[v];
        #pragma unroll
        for (int m = 8; m >= 1; m >>= 1) {
            const float ov = __shfl_xor(mv, m, 32);
            const int   oi = __shfl_xor(mi, m, 32);
            if (ov < mv || (ov == mv && oi < mi)) { mv = ov; mi = oi; }
        }
        minv[v] = mv; mini[v] = mi;
    }
    if (l16 == 0) {  // lanes 0 and 16 publish the 16 row results of this wave
        #pragma unroll
        for (int v = 0; v < 8; ++v) {
            const int row = v + half * 8;
            wgmin[row][wave] = minv[v];
            wgidx[row][wave] = mini[v];
        }
    }
    __syncthreads();

    if (tid < 16) {  // combine the 8 waves, fixed order -> deterministic
        float mv = wgmin[tid][0];
        int   mi = wgidx[tid][0];
        #pragma unroll
        for (int w = 1; w < 8; ++w) {
            const float ov = wgmin[tid][w];
            const int   oi = wgidx[tid][w];
            if (ov < mv || (ov == mv && oi < mi)) { mv = ov; mi = oi; }
        }
        fidx[tid] = mi;
    }
    __syncthreads();

    // epilogue: q = codebook[idx]; out = q + pos_emb; commit partial = sum (q-z)^2
    float ls = 0.f;
    for (int e = tid; e < 16 * DIM; e += 256) {
        const int r = e >> 9, d = e & 511;
        const int t = t0 + r;
        const int c = t & (CCH - 1);
        const float qv = cb[(size_t)fidx[r] * DIM + d];
        const float zv = zt[r * ZSTRIDE + d];
        out[(size_t)t * DIM + d] = qv + pos[c * DIM + d];
        const float df = qv - zv;
        ls += df * df;
    }
    red[tid] = ls;
    __syncthreads();
    for (int s = 128; s >= 1; s >>= 1) {
        if (tid < s) red[tid] += red[tid + s];
        __syncthreads();
    }
    if (tid == 0) partial[blockIdx.x] = red[0];
}

// ---------------- kernel 4: deterministic loss reduction ----------------
__global__ __launch_bounds__(256)
void vq_finalize(const float* __restrict__ partial, float* __restrict__ loss) {
    __shared__ float s[256];
    const int tid = threadIdx.x;
    s[tid] = partial[tid] + partial[tid + 256];
    __syncthreads();
    for (int st = 128; st >= 1; st >>= 1) {
        if (tid < st) s[tid] += s[tid + st];
        __syncthreads();
    }
    if (tid == 0) loss[0] = s[0] * (1.0f / ((float)T_TOK * (float)DIM));
}

extern "C" void kernel_launch(void* const* d_in, const int* in_sizes, int n_in,
                              void* d_out, int out_size, void* d_ws, size_t ws_size,
                              hipStream_t stream) {
    const float* x   = (const float*)d_in[0];  // [128,64,129]
    const float* W   = (const float*)d_in[1];  // [129,512]
    const float* b   = (const float*)d_in[2];  // [512]
    const float* cb  = (const float*)d_in[3];  // [8192,512]
    const float* pos = (const float*)d_in[4];  // [1,64,512]

    float* out = (float*)d_out;                // [T*D] out, then 1 loss scalar
    float* ws  = (float*)d_ws;
    float* z       = ws;                                   // 16 MB
    float* e2      = ws + (size_t)T_TOK * DIM;             // 32 KB
    float* partial = e2 + KCB;                             // 2 KB

    vq_zproj  <<<T_TOK,      128, 0, stream>>>(x, W, b, z);
    vq_cbnorm <<<KCB,        128, 0, stream>>>(cb, e2);
    vq_fused  <<<T_TOK / 16, 256, 0, stream>>>(z, cb, e2, pos, out, partial);
    vq_finalize<<<1,         256, 0, stream>>>(partial, out + (size_t)T_TOK * DIM);
}